// WarpImageLayer_36850819400525
// MI455X (gfx1250) — compile-verified
//
#include <hip/hip_runtime.h>

// Bilinear warp, border_mode = constant 0.
// image: (N,C,H,W) f32, flow: (N,2,H,W) f32 (u += flow0*W, v += flow1*H)
// out:   (N,C,H,W) f32
//
// MI455X roofline: ~201 MB total traffic @ 23.3 TB/s ~= 8.6 us floor; ~0.19
// GFLOP -> purely memory bound. No matmul structure -> WMMA inapplicable.
// gfx1250-specific choices:
//  - wave32 blocks (256 thr = 8 waves) for gather-latency hiding
//  - b128 NT loads/stores (th:TH_LOAD_NT / TH_STORE_NT) for the touch-once
//    flow + output streams, keeping the image resident in the 192 MB L2
//  - global_prefetch_b8 of the gather-center cache lines (flow sigma ~10 px,
//    so gathers land near the thread's own (y,x)); prefetch has no counter
//    cost and overlaps image first-touch latency with the flow-load chain

typedef float v4f __attribute__((ext_vector_type(4)));

#define N_  32
#define C_  3
#define H_  384
#define W_  512
#define HW_ (H_ * W_)

__global__ __launch_bounds__(256) void
warp_bilinear_kernel(const float* __restrict__ img,
                     const float* __restrict__ flow,
                     float* __restrict__ out)
{
    // Each thread produces 4 consecutive-x pixels (x all 3 channels).
    const int t  = blockIdx.x * blockDim.x + threadIdx.x;
    const int p4 = t << 2;                       // first pixel index (n,y,x flat)
    if (p4 >= N_ * HW_) return;

    const int n   = p4 / HW_;                    // const divisor -> mul/shift
    const int rem = p4 - n * HW_;                // y*W + x
    const int y   = rem / W_;
    const int x0  = rem - y * W_;

    const float* imgn = img + (size_t)n * C_ * HW_;

    // ---- speculative prefetch of the statistically-likely gather lines ----
    // (gfx1250 global_prefetch_b8: no VGPR return, no LOADcnt; speculative,
    //  so translation failures are silently dropped)
    __builtin_prefetch(imgn + rem,            0, 3);
    __builtin_prefetch(imgn + HW_ + rem,      0, 3);
    __builtin_prefetch(imgn + 2 * HW_ + rem,  0, 3);

    // ---- streamed, read-once flow: non-temporal b128 loads ----
    const float* flown = flow + (size_t)n * 2 * HW_ + rem;
    const v4f fu = __builtin_nontemporal_load((const v4f*)(flown));
    const v4f fv = __builtin_nontemporal_load((const v4f*)(flown + HW_));

    float* outn = out + (size_t)n * C_ * HW_ + rem;

    v4f o0, o1, o2;

#pragma unroll
    for (int i = 0; i < 4; ++i) {
        const float u = (float)(x0 + i) + fu[i] * (float)W_;
        const float v = (float)y        + fv[i] * (float)H_;

        const float u0f = floorf(u);
        const float v0f = floorf(v);
        const float du  = u - u0f;
        const float dv  = v - v0f;
        const int   u0  = (int)u0f;
        const int   v0  = (int)v0f;

        const bool valid = (u >= 0.0f) && (u <= (float)(W_ - 1)) &&
                           (v >= 0.0f) && (v <= (float)(H_ - 1));

        // clamp gather coordinates (reference clips each corner independently)
        const int u0c = min(max(u0,     0), W_ - 1);
        const int u1c = min(max(u0 + 1, 0), W_ - 1);
        const int v0c = min(max(v0,     0), H_ - 1);
        const int v1c = min(max(v0 + 1, 0), H_ - 1);

        // four flat offsets shared across all 3 channels
        const int r0  = v0c * W_;
        const int r1  = v1c * W_;
        const int o00 = r0 + u0c;
        const int o10 = r0 + u1c;
        const int o01 = r1 + u0c;
        const int o11 = r1 + u1c;

        const float w00 = (1.0f - du) * (1.0f - dv);
        const float w10 = du * (1.0f - dv);
        const float w01 = (1.0f - du) * dv;
        const float w11 = du * dv;

        // ---- gathers: regular-temporal loads, want L2/WGP$ residency ----
        {
            const float* ic = imgn;                  // c = 0
            float g = w00 * ic[o00] + w10 * ic[o10] + w01 * ic[o01] + w11 * ic[o11];
            o0[i] = valid ? g : 0.0f;
        }
        {
            const float* ic = imgn + HW_;            // c = 1
            float g = w00 * ic[o00] + w10 * ic[o10] + w01 * ic[o01] + w11 * ic[o11];
            o1[i] = valid ? g : 0.0f;
        }
        {
            const float* ic = imgn + 2 * HW_;        // c = 2
            float g = w00 * ic[o00] + w10 * ic[o10] + w01 * ic[o01] + w11 * ic[o11];
            o2[i] = valid ? g : 0.0f;
        }
    }

    // ---- streamed, write-once output: non-temporal b128 stores ----
    __builtin_nontemporal_store(o0, (v4f*)(outn));
    __builtin_nontemporal_store(o1, (v4f*)(outn + HW_));
    __builtin_nontemporal_store(o2, (v4f*)(outn + 2 * HW_));
}

extern "C" void kernel_launch(void* const* d_in, const int* in_sizes, int n_in,
                              void* d_out, int out_size, void* d_ws, size_t ws_size,
                              hipStream_t stream)
{
    (void)in_sizes; (void)n_in; (void)out_size; (void)d_ws; (void)ws_size;

    const float* image = (const float*)d_in[0];   // (32,3,384,512) f32
    const float* flow  = (const float*)d_in[1];   // (32,2,384,512) f32
    float*       out   = (float*)d_out;           // (32,3,384,512) f32

    const int total_pixels = N_ * HW_;            // 6,291,456
    const int threads      = total_pixels / 4;    // 1,572,864 (exact)
    const int block        = 256;                 // 8 wave32 waves
    const int grid         = threads / block;     // 6144 (exact)

    warp_bilinear_kernel<<<grid, block, 0, stream>>>(image, flow, out);
}